// DifferentiableJPEGTrigger_77919296684300
// MI455X (gfx1250) — compile-verified
//
#include <hip/hip_runtime.h>
#include <math.h>

typedef float v2f __attribute__((ext_vector_type(2)));
typedef float v8f __attribute__((ext_vector_type(8)));

#define HW 262144           // 512*512
#define NBLOCKS 1024        // 8192 patches / 8 waves per block

// Per-wave LDS ops are processed in order by hardware (DScnt in-order), so
// same-wave cross-lane store->load needs only compiler ordering, not a drain.
__device__ __forceinline__ void lds_order() {
    asm volatile("" ::: "memory");
}

__device__ __forceinline__ float clamp01(float x) {
    return fminf(fmaxf(x, 0.0f), 1.0f);
}

// Exact-to-fp32 collapse of sum_{k=-10}^{9} sigmoid(50*(x-k-0.5)) - 10.
// All terms except the nearest boundary are saturated within ~1e-11.
__device__ __forceinline__ float soft_round(float x) {
    float k = floorf(x);
    k = fminf(fmaxf(k, -10.0f), 9.0f);
    float t = 50.0f * (x - k - 0.5f);
#if __has_builtin(__builtin_amdgcn_tanhf)
    float s = __builtin_fmaf(0.5f, __builtin_amdgcn_tanhf(0.5f * t), 0.5f);
#else
    float s = __builtin_amdgcn_rcpf(1.0f + __expf(-t));
#endif
    return k + s;
}

__device__ __forceinline__ v8f wmma4(v2f a, v2f b, v8f c) {
    // (neg_a, A, neg_b, B, c_mod, C, reuse_a, reuse_b)
    return __builtin_amdgcn_wmma_f32_16x16x4_f32(false, a, false, b, (short)0, c,
                                                 false, false);
}

// A operand (16x4 chunk kb) from row-major 16x16 LDS matrix: contiguous b64.
// lane<16: M=lane, K={4kb,4kb+1}; lane>=16: M=lane-16, K={4kb+2,4kb+3}
__device__ __forceinline__ v2f loadA(const float* p, int kb, int lane) {
    int m = lane & 15;
    int k = kb * 4 + ((lane >> 4) << 1);
    return *(const v2f*)(p + m * 16 + k);
}

// B operand (4x16 chunk kb) of matrix M, read from a buffer holding M^T
// row-major: M[k][n] = M^T[n][k] -> contiguous b64.
__device__ __forceinline__ v2f loadBt(const float* pT, int kb, int lane) {
    int n = lane & 15;
    int k = kb * 4 + ((lane >> 4) << 1);
    return *(const v2f*)(pT + n * 16 + k);
}

// B operand from a buffer holding M itself row-major (strided pair).
__device__ __forceinline__ v2f loadB(const float* p, int kb, int lane) {
    int n = lane & 15;
    int k = kb * 4 + ((lane >> 4) << 1);
    v2f b;
    b.x = p[k * 16 + n];
    b.y = p[(k + 1) * 16 + n];
    return b;
}

// OUT = A @ B, A from row-major buffer, B from buffer holding B^T row-major.
__device__ __forceinline__ v8f mm16_t(const float* A, const float* BT, int lane) {
    v8f c = {};
#pragma unroll
    for (int kb = 0; kb < 4; ++kb)
        c = wmma4(loadA(A, kb, lane), loadBt(BT, kb, lane), c);
    return c;
}

// OUT = A @ B, B from row-major buffer (strided B-read). Used for stage 1 only.
__device__ __forceinline__ v8f mm16_s(const float* A, const float* B, int lane) {
    v8f c = {};
#pragma unroll
    for (int kb = 0; kb < 4; ++kb)
        c = wmma4(loadA(A, kb, lane), loadB(B, kb, lane), c);
    return c;
}

// Store D-layout regs (S) row-major: column scatter (paired b32 stores).
__device__ __forceinline__ void storeD(float* p, v8f c, int lane) {
    int n = lane & 15;
    int mb = (lane >> 4) << 3;
#pragma unroll
    for (int v = 0; v < 8; ++v)
        p[(mb + v) * 16 + n] = c[v];
}

// Store D-layout regs TRANSPOSED (buffer gets S^T row-major): 2 x ds_store_b128.
__device__ __forceinline__ void storeDT(float* p, v8f c, int lane) {
    int n = lane & 15;
    int mb = (lane >> 4) << 3;
    float* q = p + n * 16 + mb;
    *(float4*)(q) = make_float4(c[0], c[1], c[2], c[3]);
    *(float4*)(q + 4) = make_float4(c[4], c[5], c[6], c[7]);
}

__global__ __launch_bounds__(256) void diff_jpeg_kernel(
    const float* __restrict__ img, const float* __restrict__ qy,
    const float* __restrict__ qc, float* __restrict__ out) {
    __shared__ float sDD[256];          // diag(D,D) row-major  (= DDT^T)
    __shared__ float sDDT[256];         // diag(D,D)^T row-major (= DD^T)
    __shared__ float2 sQP[2][64];       // {clipped q, 1/q} luma / chroma
    __shared__ float sTile[8][4][256];  // per wave: Y, Cb, Cr, temp

    const int t = threadIdx.x;
    const int lane = t & 31;
    const int wave = t >> 5;

    // Build constants (one element per thread)
    {
        int i = t >> 4, j = t & 15;
        float val = 0.0f;
        if ((i >> 3) == (j >> 3)) {
            int m = i & 7, n = j & 7;
            val = (m == 0) ? 0.35355339059327373f
                           : 0.5f * __cosf((float)((2 * n + 1) * m) *
                                           0.19634954084936207f);  // pi/16
        }
        sDD[i * 16 + j] = val;
        sDDT[j * 16 + i] = val;
        if (t < 128) {
            float q = (t < 64) ? qy[t] : qc[t - 64];
            q = fminf(fmaxf(q, 2.0f), 15.0f);
            sQP[t >> 6][t & 63] = make_float2(q, 1.0f / q);
        }
    }
    __syncthreads();

    // One 16x16 patch (= four 8x8 JPEG blocks, all 3 channels) per wave
    const int gw = blockIdx.x * 8 + wave;   // 0..8191
    const int b = gw >> 10;                 // image
    const int pr = (gw >> 5) & 31;          // patch row
    const int pc = gw & 31;                 // patch col
    const int base = b * 3 * HW + (pr * 16) * 512 + pc * 16;

    float* tY = sTile[wave][0];
    float* tCb = sTile[wave][1];
    float* tCr = sTile[wave][2];
    float* tT = sTile[wave][3];

    // Load RGB patch as float4 (b128, coalesced), convert to YCbCr tiles in LDS
    const int row0 = lane >> 2;             // 0..7
    const int c4 = (lane & 3) << 2;         // 0,4,8,12
#pragma unroll
    for (int i = 0; i < 2; ++i) {
        int row = row0 + 8 * i;
        int off = base + row * 512 + c4;
        float4 R = *(const float4*)(img + off);
        float4 G = *(const float4*)(img + off + HW);
        float4 B4 = *(const float4*)(img + off + 2 * HW);
        float4 Yv, Cbv, Crv;
        const float* rp = (const float*)&R;
        const float* gp = (const float*)&G;
        const float* bp = (const float*)&B4;
        float* yp = (float*)&Yv;
        float* cbp = (float*)&Cbv;
        float* crp = (float*)&Crv;
#pragma unroll
        for (int c = 0; c < 4; ++c) {
            float r = clamp01(rp[c]), g = clamp01(gp[c]), bl = clamp01(bp[c]);
            yp[c] = 0.299f * r + 0.587f * g + 0.114f * bl;
            cbp[c] = -0.168736f * r - 0.331264f * g + 0.5f * bl + 0.5f;
            crp[c] = 0.5f * r - 0.418688f * g - 0.081312f * bl + 0.5f;
        }
        *(float4*)(tY + row * 16 + c4) = Yv;
        *(float4*)(tCb + row * 16 + c4) = Cbv;
        *(float4*)(tCr + row * 16 + c4) = Crv;
    }
    lds_order();

#pragma unroll
    for (int ch = 0; ch < 3; ++ch) {
        float* tX = sTile[wave][ch];
        const float2* qp = sQP[ch == 0 ? 0 : 1];

        // Stage 1: T1 = DD @ X   (A=sDD contig; B=X strided)
        v8f acc = mm16_s(sDD, tX, lane);
        storeD(tT, acc, lane);              // tT := T1 row-major
        lds_order();

        // Stage 2: coef = T1 @ DD  (A=T1 contig; B=DD via sDDT contig)
        acc = mm16_t(tT, sDDT, lane);

        // quant = soft_round(coef * 1/q) * q ; m&7 == v, n&7 == lane&7
#pragma unroll
        for (int v = 0; v < 8; ++v) {
            float2 q2 = qp[v * 8 + (lane & 7)];
            acc[v] = soft_round(acc[v] * q2.y) * q2.x;
        }
        storeDT(tT, acc, lane);             // tT := Q^T row-major (2xb128)
        lds_order();

        // Stage 3: W = Q^T @ DD = (DDT @ Q)^T = T3^T  (A=Q^T contig; B contig)
        acc = mm16_t(tT, sDDT, lane);
        storeDT(tT, acc, lane);             // tT := (T3^T)^T = T3 row-major
        lds_order();

        // Stage 4: rec = T3 @ DDT  (A=T3 contig; B=DDT via sDD contig)
        acc = mm16_t(tT, sDD, lane);
        storeD(tX, acc, lane);              // channel tile := rec row-major
        lds_order();
    }

    // YCbCr -> RGB, clip, store as float4 (b128, coalesced)
#pragma unroll
    for (int i = 0; i < 2; ++i) {
        int row = row0 + 8 * i;
        int off = base + row * 512 + c4;
        float4 Yv = *(const float4*)(tY + row * 16 + c4);
        float4 Cbv = *(const float4*)(tCb + row * 16 + c4);
        float4 Crv = *(const float4*)(tCr + row * 16 + c4);
        float4 Ro, Go, Bo;
        const float* yp = (const float*)&Yv;
        const float* cbp = (const float*)&Cbv;
        const float* crp = (const float*)&Crv;
        float* rop = (float*)&Ro;
        float* gop = (float*)&Go;
        float* bop = (float*)&Bo;
#pragma unroll
        for (int c = 0; c < 4; ++c) {
            float y2 = yp[c];
            float cb2 = cbp[c] - 0.5f;
            float cr2 = crp[c] - 0.5f;
            rop[c] = clamp01(y2 + 1.402f * cr2);
            gop[c] = clamp01(y2 - 0.344136f * cb2 - 0.714136f * cr2);
            bop[c] = clamp01(y2 + 1.772f * cb2);
        }
        *(float4*)(out + off) = Ro;
        *(float4*)(out + off + HW) = Go;
        *(float4*)(out + off + 2 * HW) = Bo;
    }
}

extern "C" void kernel_launch(void* const* d_in, const int* in_sizes, int n_in,
                              void* d_out, int out_size, void* d_ws, size_t ws_size,
                              hipStream_t stream) {
    const float* img = (const float*)d_in[0];
    const float* qy = (const float*)d_in[1];
    const float* qc = (const float*)d_in[2];
    float* out = (float*)d_out;
    diff_jpeg_kernel<<<NBLOCKS, 256, 0, stream>>>(img, qy, qc, out);
}